// HyperGraphStructuralLayer_Knn_and_Kmeans_19825569038842
// MI455X (gfx1250) — compile-verified
//
#include <hip/hip_runtime.h>

typedef float v2f __attribute__((ext_vector_type(2)));
typedef float v8f __attribute__((ext_vector_type(8)));

#define F_DIM 256

// ---------------- utility: zero fill ----------------
__global__ __launch_bounds__(256) void fill_zero(float* __restrict__ p, int n) {
    int i = blockIdx.x * blockDim.x + threadIdx.x;
    if (i < n) p[i] = 0.0f;
}

// ---------------- degree accumulation ----------------
__global__ __launch_bounds__(256) void degree_kernel(const int* __restrict__ src,
                                                     const int* __restrict__ dst,
                                                     float* __restrict__ D,
                                                     float* __restrict__ B, int nnz) {
    int e = blockIdx.x * blockDim.x + threadIdx.x;
    if (e >= nnz) return;
    atomicAdd(&D[src[e]], 1.0f);
    atomicAdd(&B[dst[e]], 1.0f);
}

__global__ __launch_bounds__(256) void invert_kernel(float* __restrict__ p, int n) {
    int i = blockIdx.x * blockDim.x + threadIdx.x;
    if (i >= n) return;
    float v = p[i];
    p[i] = v > 0.0f ? 1.0f / v : 0.0f;
}

// ---------------- WMMA f32 GEMM: Y[M,256] = X[M,256] @ W[256,256] ----------------
// One wave per (16-row strip, 128-col half). 8 accumulator tiles, A-frag reused 8x.
__global__ __launch_bounds__(256) void gemm_wmma_f32(const float* __restrict__ X,
                                                     const float* __restrict__ W,
                                                     float* __restrict__ Y, int M) {
    const int wave = threadIdx.x >> 5;
    const int lane = threadIdx.x & 31;
    const int job  = blockIdx.x * 8 + wave;     // wave-level job id
    const int strips = M >> 4;                  // M divisible by 16 (100000/16 = 6250)
    const int strip = job >> 1;
    const int nhalf = job & 1;
    if (strip >= strips) return;                // wave-uniform: EXEC stays all-ones

    const int m0    = strip << 4;
    const int n0    = nhalf << 7;               // 0 or 128
    const int mrow  = m0 + (lane & 15);
    const int khalf = lane >> 4;                // 0 or 1
    const int ncol  = lane & 15;

    v8f acc[8] = {};

    const float* arow = X + (size_t)mrow * F_DIM;

    for (int k = 0; k < F_DIM; k += 4) {
        // A fragment: V0 = X[m][k + 2*khalf], V1 = X[m][k + 2*khalf + 1]
        v2f a;
        a.x = arow[k + 2 * khalf];
        a.y = arow[k + 2 * khalf + 1];
#pragma unroll
        for (int t = 0; t < 8; ++t) {
            int n = n0 + (t << 4) + ncol;
            v2f b;
            b.x = W[(size_t)(k + 2 * khalf) * F_DIM + n];
            b.y = W[(size_t)(k + 2 * khalf + 1) * F_DIM + n];
            acc[t] = __builtin_amdgcn_wmma_f32_16x16x4_f32(
                false, a, false, b, (short)0, acc[t], false, false);
        }
    }

    // C/D layout: VGPR r -> row m0+r (lanes 0-15) / m0+8+r (lanes 16-31), col = lane%16
#pragma unroll
    for (int t = 0; t < 8; ++t) {
        int n = n0 + (t << 4) + ncol;
#pragma unroll
        for (int r = 0; r < 8; ++r) {
            int row = m0 + r + (khalf << 3);
            Y[(size_t)row * F_DIM + n] = acc[t][r];
        }
    }
}

// ---------------- edge propagation: out[sidx[e]] += rows[gidx[e]] * scale(gidx[e]) ----
// One wave per edge; each lane moves 8 floats (two float4). Atomic targets are
// 100 MB -> resident in the 192 MB L2, so atomics never hit HBM.
__global__ __launch_bounds__(256) void scatter_rows(const float* __restrict__ rows,
                                                    const int* __restrict__ gidx,
                                                    const int* __restrict__ sidx,
                                                    const float* __restrict__ gscale,
                                                    float* __restrict__ out, int nnz) {
    int e = blockIdx.x * 8 + (threadIdx.x >> 5);
    if (e >= nnz) return;
    int lane = threadIdx.x & 31;
    int g = gidx[e];
    int s = sidx[e];
    float scale = gscale ? gscale[g] : 1.0f;

    const float4* rp = (const float4*)(rows + (size_t)g * F_DIM) + lane;
    float4 v0 = rp[0];    // floats [4*lane, 4*lane+4)
    float4 v1 = rp[32];   // floats [128 + 4*lane, ...)

    float* op = out + (size_t)s * F_DIM;
    int o0 = lane << 2;
    int o1 = 128 + (lane << 2);
    atomicAdd(op + o0 + 0, v0.x * scale);
    atomicAdd(op + o0 + 1, v0.y * scale);
    atomicAdd(op + o0 + 2, v0.z * scale);
    atomicAdd(op + o0 + 3, v0.w * scale);
    atomicAdd(op + o1 + 0, v1.x * scale);
    atomicAdd(op + o1 + 1, v1.y * scale);
    atomicAdd(op + o1 + 2, v1.z * scale);
    atomicAdd(op + o1 + 3, v1.w * scale);
}

// ---------------- epilogue: out = prelu(acc*Dinv[row] + bias[col] (+ residual)) -------
__global__ __launch_bounds__(256) void epilogue_kernel(const float* __restrict__ acc,
                                                       const float* __restrict__ Dinv,
                                                       const float* __restrict__ bias,
                                                       const float* __restrict__ aP,
                                                       const float* __restrict__ residual,
                                                       float* __restrict__ out, int total) {
    int i = blockIdx.x * blockDim.x + threadIdx.x;
    if (i >= total) return;
    int row = i >> 8;
    int col = i & (F_DIM - 1);
    float v = acc[i] * Dinv[row] + bias[col];
    if (residual) v += residual[i];
    float a = aP[0];
    out[i] = v >= 0.0f ? v : a * v;
}

extern "C" void kernel_launch(void* const* d_in, const int* in_sizes, int n_in,
                              void* d_out, int out_size, void* d_ws, size_t ws_size,
                              hipStream_t stream) {
    const float* x  = (const float*)d_in[0];
    const int*   ei = (const int*)d_in[1];
    const float* W1 = (const float*)d_in[2];
    const float* b1 = (const float*)d_in[3];
    const float* W2 = (const float*)d_in[4];
    const float* b2 = (const float*)d_in[5];
    const float* aP = (const float*)d_in[6];

    const int M   = in_sizes[0] / F_DIM;   // 100000
    const int nnz = in_sizes[1] / 2;       // 1600000
    const int* src = ei;
    const int* dst = ei + nnz;

    const size_t MF = (size_t)M * F_DIM;
    float* bufA = (float*)d_ws;            // xt / acc ping-pong
    float* bufB = bufA + MF;               // m   / h   ping-pong
    float* Dd   = bufB + MF;               // node degree  -> Dinv (in place)
    float* Bd   = Dd + M;                  // hedge degree -> Binv (in place)

    const int total = (int)MF;
    const dim3 blk(256);
    const int gFill   = (total + 255) / 256;
    const int gSmall  = (2 * M + 255) / 256;
    const int gEdge   = (nnz + 255) / 256;
    const int gScat   = (nnz + 7) / 8;           // 8 edges per 256-thread block
    const int jobs    = (M / 16) * 2;
    const int gGemm   = (jobs + 7) / 8;          // 8 waves per block
    float* outF = (float*)d_out;

    // --- degrees (shared by both layers), inverted in place ---
    fill_zero<<<gSmall, blk, 0, stream>>>(Dd, 2 * M);
    degree_kernel<<<gEdge, blk, 0, stream>>>(src, dst, Dd, Bd, nnz);
    invert_kernel<<<gSmall, blk, 0, stream>>>(Dd, 2 * M);   // Dd,Bd contiguous

    // ================= Layer 1 =================
    gemm_wmma_f32<<<gGemm, blk, 0, stream>>>(x, W1, bufA, M);              // xt
    fill_zero<<<gFill, blk, 0, stream>>>(bufB, total);
    scatter_rows<<<gScat, blk, 0, stream>>>(bufA, src, dst, nullptr, bufB, nnz); // m_raw
    fill_zero<<<gFill, blk, 0, stream>>>(bufA, total);
    scatter_rows<<<gScat, blk, 0, stream>>>(bufB, dst, src, Bd, bufA, nnz);      // acc (Binv folded)
    epilogue_kernel<<<gFill, blk, 0, stream>>>(bufA, Dd, b1, aP, nullptr, bufB, total); // h = prelu(...)

    // ================= Layer 2 =================
    gemm_wmma_f32<<<gGemm, blk, 0, stream>>>(bufB, W2, bufA, M);           // xt2
    fill_zero<<<gFill, blk, 0, stream>>>(bufB, total);
    scatter_rows<<<gScat, blk, 0, stream>>>(bufA, src, dst, nullptr, bufB, nnz);
    fill_zero<<<gFill, blk, 0, stream>>>(bufA, total);
    scatter_rows<<<gScat, blk, 0, stream>>>(bufB, dst, src, Bd, bufA, nnz);
    epilogue_kernel<<<gFill, blk, 0, stream>>>(bufA, Dd, b2, aP, x, outF, total);  // +residual, prelu
}